// MultiHeadAttention_88545045775231
// MI455X (gfx1250) — compile-verified
//
#include <hip/hip_runtime.h>

typedef __attribute__((ext_vector_type(2))) float v2f;
typedef __attribute__((ext_vector_type(4))) float v4f;
typedef __attribute__((ext_vector_type(8))) float v8f;
typedef int v4i_a __attribute__((vector_size(16)));   // matches builtin's b128 pointee

#define DM    1024
#define NH    16
#define DEPTH 64
#define BB    2
#define SS    2048
#define BS    (BB * SS)         // 4096 rows total

#define TILE_M 64
#define TILE_N 64
#define TILE_K 16
#define PITCH  20               // LDS row pitch in floats: 80B => 16B-aligned rows

#ifndef __has_builtin
#define __has_builtin(x) 0
#endif

#if __has_builtin(__builtin_amdgcn_global_load_async_to_lds_b128)
#define HAVE_ASYNC 1
#else
#define HAVE_ASYNC 0
#endif

#if HAVE_ASYNC
#define ASYNC_B128(gp, lp)                                                     \
    __builtin_amdgcn_global_load_async_to_lds_b128(                            \
        (__attribute__((address_space(1))) v4i_a*)(gp),                        \
        (__attribute__((address_space(3))) v4i_a*)(lp), 0, 0)

__device__ __forceinline__ void wait_async0() {
#if __has_builtin(__builtin_amdgcn_s_wait_asynccnt)
    __builtin_amdgcn_s_wait_asynccnt(0);
#else
    asm volatile("s_wait_asynccnt 0x0" ::: "memory");
#endif
}
#endif

// Generic fp32 WMMA GEMM, double-buffered LDS staging:
//   C[bo,bi] = alpha * A[bo,bi] (x) B[bo,bi]  (+ bias[n]) (+ mask[bo][n] * -1e9)
// batch z = bo*innerB + bi;  per-operand (outer,inner) batch strides in elements.
// transB: B element [k][n] lives at Bp[n*ldb + k] (memory is n-major, k-contig).
__launch_bounds__(128)
__global__ void gemm_f32_wmma(
    const float* __restrict__ A, long long sAo, long long sAi, int lda,
    const float* __restrict__ B, long long sBo, long long sBi, int ldb, int transB,
    float*       __restrict__ C, long long sCo, long long sCi, int ldc,
    const float* __restrict__ bias,
    const float* __restrict__ mask, long long maskStride,
    int innerB, float alpha, int K)
{
    __shared__ float As[2][TILE_M * PITCH];
    __shared__ float Bt[2][TILE_N * PITCH];  // stored transposed: [n][k]

    const int tid  = threadIdx.x;
    const int wave = tid >> 5;
    const int lane = tid & 31;
    const int m16  = lane & 15;
    const int hi   = lane >> 4;              // which K-half / M-half per ISA layout

    const int bz = blockIdx.z;
    const int bo = bz / innerB;
    const int bi = bz - bo * innerB;

    const float* Ap = A + (long long)bo * sAo + (long long)bi * sAi;
    const float* Bp = B + (long long)bo * sBo + (long long)bi * sBi;
    float*       Cp = C + (long long)bo * sCo + (long long)bi * sCi;
    const float* maskRow = mask ? (mask + (long long)bo * maskStride) : nullptr;

    const int m0 = blockIdx.y * TILE_M;
    const int n0 = blockIdx.x * TILE_N;

    // ---- per-thread staging coordinates (256 float4 chunks per tile) ----
    const int f[2]    = { tid, tid + 128 };
    int        aRow[2], aC4[2], aL[2];        // A tile 64x16, row-major
    const float* aG[2];
    int        bLt[2];                        // transB: B tile 64(n) x 16(k), k-contig
    const float* bGt[2];
    int        bKK[2], bN4[2];                // !transB: B tile 16(k) x 64(n)
    const float* bGn[2];
#pragma unroll
    for (int it = 0; it < 2; ++it) {
        aRow[it] = f[it] >> 2;
        aC4[it]  = (f[it] & 3) << 2;
        aL[it]   = aRow[it] * PITCH + aC4[it];
        aG[it]   = Ap + (long long)(m0 + aRow[it]) * lda + aC4[it];

        bLt[it]  = aL[it];
        bGt[it]  = Bp + (long long)(n0 + aRow[it]) * ldb + aC4[it];

        bKK[it]  = f[it] >> 4;
        bN4[it]  = (f[it] & 15) << 2;
        bGn[it]  = Bp + (long long)bKK[it] * ldb + n0 + bN4[it];
    }

    v4f ra[2], rb[2];

    // issue stage: start loads for tile at k0 into LDS buffer b
    auto stage_issue = [&](int k0, int b) {
#if HAVE_ASYNC
#pragma unroll
        for (int it = 0; it < 2; ++it) {
            ASYNC_B128(aG[it] + k0, &As[b][aL[it]]);
            __builtin_prefetch(aG[it] + k0 + TILE_K, 0, 3);
        }
        if (transB) {
#pragma unroll
            for (int it = 0; it < 2; ++it)
                ASYNC_B128(bGt[it] + k0, &Bt[b][bLt[it]]);
        } else {
#pragma unroll
            for (int it = 0; it < 2; ++it)
                rb[it] = *(const v4f*)(bGn[it] + (long long)k0 * ldb);
        }
#else
#pragma unroll
        for (int it = 0; it < 2; ++it) {
            ra[it] = *(const v4f*)(aG[it] + k0);
            __builtin_prefetch(aG[it] + k0 + TILE_K, 0, 3);
        }
        if (transB) {
#pragma unroll
            for (int it = 0; it < 2; ++it)
                rb[it] = *(const v4f*)(bGt[it] + k0);
        } else {
#pragma unroll
            for (int it = 0; it < 2; ++it)
                rb[it] = *(const v4f*)(bGn[it] + (long long)k0 * ldb);
        }
#endif
    };

    // commit stage: finish writing tile into LDS buffer b (sync paths only)
    auto stage_commit = [&](int b) {
#if !HAVE_ASYNC
#pragma unroll
        for (int it = 0; it < 2; ++it)
            *(v4f*)&As[b][aL[it]] = ra[it];
        if (transB) {
#pragma unroll
            for (int it = 0; it < 2; ++it)
                *(v4f*)&Bt[b][bLt[it]] = rb[it];
        }
#endif
        if (!transB) {
#pragma unroll
            for (int it = 0; it < 2; ++it) {
                Bt[b][(bN4[it] + 0) * PITCH + bKK[it]] = rb[it].x;
                Bt[b][(bN4[it] + 1) * PITCH + bKK[it]] = rb[it].y;
                Bt[b][(bN4[it] + 2) * PITCH + bKK[it]] = rb[it].z;
                Bt[b][(bN4[it] + 3) * PITCH + bKK[it]] = rb[it].w;
            }
        }
    };

    v8f acc[4] = {};   // wave's 16x64 strip = 4 tiles of 16x16

    // ---- prologue: stage tile 0 into buffer 0 ----
    stage_issue(0, 0);
    stage_commit(0);

    int cur = 0;
    for (int k0 = 0; k0 < K; k0 += TILE_K) {
        const bool hasNext = (k0 + TILE_K) < K;
#if HAVE_ASYNC
        wait_async0();                // my tile-k async writes have landed in LDS
#endif
        __syncthreads();              // everyone's tile-k data visible; buf[nxt] free
        const int nxt = cur ^ 1;
        if (hasNext) stage_issue(k0 + TILE_K, nxt);

        // ---- 16 x V_WMMA_F32_16X16X4_F32 on tile k ----
        const float* Acur = As[cur];
        const float* Bcur = Bt[cur];
#pragma unroll
        for (int k4 = 0; k4 < 4; ++k4) {
            int kk = (k4 << 2) + (hi << 1);       // A frag: a[j] = A[m][kk+j]
            v2f a = *(const v2f*)&Acur[(wave * 16 + m16) * PITCH + kk];
#pragma unroll
            for (int nt = 0; nt < 4; ++nt) {
                v2f b = *(const v2f*)&Bcur[(nt * 16 + m16) * PITCH + kk]; // b[j]=B[kk+j][n]
                acc[nt] = __builtin_amdgcn_wmma_f32_16x16x4_f32(
                    false, a, false, b, (short)0, acc[nt], false, false);
            }
        }

        if (hasNext) stage_commit(nxt);
        cur = nxt;
    }

    // ---- epilogue: D layout -> C[m = r + 8*hi][n = lane&15] per VGPR r ----
#pragma unroll
    for (int nt = 0; nt < 4; ++nt) {
        int gn = n0 + nt * 16 + m16;
        float add = (bias ? bias[gn] : 0.0f) + (maskRow ? maskRow[gn] * -1e9f : 0.0f);
#pragma unroll
        for (int r = 0; r < 8; ++r) {
            int gm = m0 + wave * 16 + (hi << 3) + r;
            Cp[(long long)gm * ldc + gn] = alpha * acc[nt][r] + add;
        }
    }
}

// In-place row softmax over 2048 columns; one 256-thread block per row.
__launch_bounds__(256)
__global__ void softmax_rows_2048(float* __restrict__ W)
{
    __shared__ float red[256];
    float* p = W + (long long)blockIdx.x * 2048;
    const int tid = threadIdx.x;

    float x[8];
    float mx = -INFINITY;
#pragma unroll
    for (int i = 0; i < 8; ++i) {
        x[i] = p[tid + (i << 8)];
        mx = fmaxf(mx, x[i]);
    }
    red[tid] = mx; __syncthreads();
    for (int s = 128; s > 0; s >>= 1) {
        if (tid < s) red[tid] = fmaxf(red[tid], red[tid + s]);
        __syncthreads();
    }
    mx = red[0]; __syncthreads();

    float sum = 0.0f;
#pragma unroll
    for (int i = 0; i < 8; ++i) { x[i] = __expf(x[i] - mx); sum += x[i]; }
    red[tid] = sum; __syncthreads();
    for (int s = 128; s > 0; s >>= 1) {
        if (tid < s) red[tid] += red[tid + s];
        __syncthreads();
    }
    float inv = 1.0f / red[0];
#pragma unroll
    for (int i = 0; i < 8; ++i) p[tid + (i << 8)] = x[i] * inv;
}

extern "C" void kernel_launch(void* const* d_in, const int* in_sizes, int n_in,
                              void* d_out, int out_size, void* d_ws, size_t ws_size,
                              hipStream_t stream) {
    (void)in_sizes; (void)n_in; (void)out_size; (void)ws_size;

    const float* Q    = (const float*)d_in[0];
    const float* Kin  = (const float*)d_in[1];
    const float* V    = (const float*)d_in[2];
    const float* mask = (const float*)d_in[3];
    const float* Wq   = (const float*)d_in[4];
    const float* bq   = (const float*)d_in[5];
    const float* Wk   = (const float*)d_in[6];
    const float* bk   = (const float*)d_in[7];
    const float* Wv   = (const float*)d_in[8];
    const float* bv   = (const float*)d_in[9];
    const float* Wo   = (const float*)d_in[10];
    const float* bo   = (const float*)d_in[11];

    float* out     = (float*)d_out;                       // (B,S,DM)
    float* weights = out + (long long)BS * DM;            // (B,H,S,S)

    float* qbuf = (float*)d_ws;                           // (B,S,DM)
    float* kbuf = qbuf + (long long)BS * DM;
    float* vbuf = kbuf + (long long)BS * DM;
    float* cbuf = vbuf + (long long)BS * DM;              // concat heads (B,S,DM)

    const dim3 blk(128);
    const float scale = 0.125f;                            // 1/sqrt(64)

    // QKV projections: (4096x1024)@(1024x1024) + bias
    gemm_f32_wmma<<<dim3(DM / TILE_N, BS / TILE_M, 1), blk, 0, stream>>>(
        Q,   0, 0, DM,  Wq, 0, 0, DM, 0,  qbuf, 0, 0, DM,  bq, nullptr, 0, 1, 1.0f, DM);
    gemm_f32_wmma<<<dim3(DM / TILE_N, BS / TILE_M, 1), blk, 0, stream>>>(
        Kin, 0, 0, DM,  Wk, 0, 0, DM, 0,  kbuf, 0, 0, DM,  bk, nullptr, 0, 1, 1.0f, DM);
    gemm_f32_wmma<<<dim3(DM / TILE_N, BS / TILE_M, 1), blk, 0, stream>>>(
        V,   0, 0, DM,  Wv, 0, 0, DM, 0,  vbuf, 0, 0, DM,  bv, nullptr, 0, 1, 1.0f, DM);

    // scores = scale * q @ k^T + mask*-1e9   per (b,h): M=N=2048, K=64
    gemm_f32_wmma<<<dim3(SS / TILE_N, SS / TILE_M, BB * NH), blk, 0, stream>>>(
        qbuf, (long long)SS * DM, DEPTH, DM,
        kbuf, (long long)SS * DM, DEPTH, DM, 1 /*transB: k^T*/,
        weights, (long long)NH * SS * SS, (long long)SS * SS, SS,
        nullptr, mask, SS, NH, scale, DEPTH);

    // softmax over key dim, in place (B*H*S rows of 2048)
    softmax_rows_2048<<<dim3(BB * NH * SS), dim3(256), 0, stream>>>(weights);

    // ctx = weights @ v   per (b,h): M=2048, N=64, K=2048 -> concat layout in cbuf
    gemm_f32_wmma<<<dim3(DEPTH / TILE_N, SS / TILE_M, BB * NH), blk, 0, stream>>>(
        weights, (long long)NH * SS * SS, (long long)SS * SS, SS,
        vbuf, (long long)SS * DM, DEPTH, DM, 0,
        cbuf, (long long)SS * DM, DEPTH, DM,
        nullptr, nullptr, 0, NH, 1.0f, SS);

    // output = concat @ Wo + bo
    gemm_f32_wmma<<<dim3(DM / TILE_N, BS / TILE_M, 1), blk, 0, stream>>>(
        cbuf, 0, 0, DM,  Wo, 0, 0, DM, 0,  out, 0, 0, DM,  bo, nullptr, 0, 1, 1.0f, DM);
}